// DTWLoss_91173565760107
// MI455X (gfx1250) — compile-verified
//
#include <hip/hip_runtime.h>
#include <hip/hip_bf16.h>
#include <stdint.h>

// DTW loss, B=64, T=1024, F=4 (first 2 coords used).
// d_ws layout: [0, 16MB): packed 2-bit direction matrices (64 x 65536 u32)
//              [16MB, 16MB+256): per-batch losses (64 f32)

#define Tn 1024
#define Bn 64
#define INFV 1e30f
#define WORDS_PER_BATCH (Tn * Tn / 16)   // 65536 u32 = 256 KB packed

__global__ __launch_bounds__(1024)
void dtw_forward_kernel(const float* __restrict__ preds,
                        const float* __restrict__ targs,
                        unsigned* __restrict__ dirP)
{
    __shared__ float q2[2 * Tn];     // interleaved (x,y) of targ points
    __shared__ float diag[3 * Tn];   // rolling anti-diagonals of C

    const int b = blockIdx.x;        // batch
    const int i = threadIdx.x;       // this thread owns DP row i

    const size_t rowBase = ((size_t)b * Tn + i) * 4;  // F == 4

    // ---- CDNA5 async global->LDS staging of q = targs[b, i, 0:2] ----
    {
        unsigned lds_off = (unsigned)(uintptr_t)(&q2[2 * i]);
        unsigned long long ga = (unsigned long long)(uintptr_t)(targs + rowBase);
        asm volatile("global_load_async_to_lds_b64 %0, %1, off"
                     :: "v"(lds_off), "v"(ga) : "memory");
        asm volatile("s_wait_asynccnt 0x0" ::: "memory");
    }

    const float px = preds[rowBase + 0];
    const float py = preds[rowBase + 1];

    __syncthreads();  // all waves' async copies visible in LDS

    unsigned* __restrict__ dirB = dirP + (size_t)b * WORDS_PER_BATCH;

    unsigned o0 = 0, o1 = Tn, o2 = 2 * Tn;   // cur, prev1, prev2 offsets
    unsigned w = 0;                          // 16 packed 2-bit directions

    for (int s = 0; s < 2 * Tn - 1; ++s) {
        const int j = s - i;                 // sweeps 0..1023 contiguously
        if (j >= 0 && j < Tn) {
            const float cu = (i > 0)          ? diag[o1 + i - 1] : INFV; // C[i-1,j]
            const float cl = (j > 0)          ? diag[o1 + i]     : INFV; // C[i,j-1]
            const float cd = (i > 0 && j > 0) ? diag[o2 + i - 1] : INFV; // C[i-1,j-1]

            // argmin([cd, cu, cl]) with first-min tie-break (matches jnp.argmin)
            float m; unsigned k;
            if (cd <= cu && cd <= cl) { m = cd; k = 0u; }
            else if (cu <= cl)        { m = cu; k = 1u; }
            else                      { m = cl; k = 2u; }
            if (i == 0 && j == 0) m = 0.0f;   // DP origin: C[0,0] = D[0,0]

            const float dx = px - q2[2 * j + 0];
            const float dy = py - q2[2 * j + 1];
            const float d  = sqrtf(dx * dx + dy * dy);

            diag[o0 + i] = d + m;

            // pack within-thread: j advances by 1 per step for fixed row i
            const unsigned sh = 2u * (unsigned)(j & 15);
            w = ((j & 15) == 0) ? k : (w | (k << sh));
            if ((j & 15) == 15)
                dirB[(unsigned)i * (Tn / 16) + (unsigned)(j >> 4)] = w;
        }
        __syncthreads();
        const unsigned t = o2; o2 = o1; o1 = o0; o0 = t;  // rotate diagonals
    }
}

__global__ __launch_bounds__(1024)
void dtw_backtrace_kernel(const float* __restrict__ preds,
                          const float* __restrict__ targs,
                          const float* __restrict__ subcoef,
                          const unsigned* __restrict__ dirP,
                          float* __restrict__ lossOut)
{
    // Dynamic LDS: 65536 u32 packed dirs (256KB) + 2048 f32 p + 2048 f32 t
    extern __shared__ unsigned lds[];
    unsigned* dirL = lds;
    float* p2 = (float*)(lds + WORDS_PER_BATCH);
    float* t2 = p2 + 2 * Tn;

    const int b   = blockIdx.x;
    const int tid = threadIdx.x;

    // ---- async-stage this batch's packed direction matrix into LDS ----
    {
        const unsigned* g = dirP + (size_t)b * WORDS_PER_BATCH + tid * 4;
        const unsigned lbase = (unsigned)(uintptr_t)(dirL + tid * 4);
        #pragma unroll
        for (int c = 0; c < 16; ++c) {           // 16 x (1024 thr x 16B) = 256KB
            unsigned ldsoff = lbase + c * 16384;
            unsigned long long ga = (unsigned long long)(uintptr_t)(g + c * 4096);
            asm volatile("global_load_async_to_lds_b128 %0, %1, off"
                         :: "v"(ldsoff), "v"(ga) : "memory");
        }
    }
    // ---- async-stage p[:, 0:2] and t[:, 0:2] into LDS ----
    {
        const size_t rowBase = ((size_t)b * Tn + tid) * 4;
        unsigned po = (unsigned)(uintptr_t)(&p2[2 * tid]);
        unsigned long long gp = (unsigned long long)(uintptr_t)(preds + rowBase);
        asm volatile("global_load_async_to_lds_b64 %0, %1, off"
                     :: "v"(po), "v"(gp) : "memory");
        unsigned to = (unsigned)(uintptr_t)(&t2[2 * tid]);
        unsigned long long gt = (unsigned long long)(uintptr_t)(targs + rowBase);
        asm volatile("global_load_async_to_lds_b64 %0, %1, off"
                     :: "v"(to), "v"(gt) : "memory");
    }
    asm volatile("s_wait_asynccnt 0x0" ::: "memory");
    __syncthreads();

    if (tid == 0) {
        const float sc0 = subcoef[0];
        const float sc1 = subcoef[1];
        int i = Tn - 1, j = Tn - 1;
        float loss = 0.0f;
        for (;;) {
            loss += fabsf(p2[2 * i + 0] - t2[2 * j + 0]) * sc0
                  + fabsf(p2[2 * i + 1] - t2[2 * j + 1]) * sc1;
            if (i == 0 && j == 0) break;
            const unsigned wv = dirL[(unsigned)i * (Tn / 16) + (unsigned)(j >> 4)];
            const unsigned k  = (wv >> (2u * (unsigned)(j & 15))) & 3u;
            if (k == 0u)      { --i; --j; }
            else if (k == 1u) { --i; }
            else              { --j; }
        }
        lossOut[b] = loss;
    }
}

__global__ void dtw_reduce_kernel(const float* __restrict__ lossOut,
                                  float* __restrict__ out)
{
    float s = 0.0f;
    for (int b = 0; b < Bn; ++b) s += lossOut[b];   // fixed order: deterministic
    out[0] = s;
}

extern "C" void kernel_launch(void* const* d_in, const int* in_sizes, int n_in,
                              void* d_out, int out_size, void* d_ws, size_t ws_size,
                              hipStream_t stream)
{
    (void)in_sizes; (void)n_in; (void)out_size; (void)ws_size;
    const float* preds   = (const float*)d_in[0];  // [64,1024,4] f32
    const float* targs   = (const float*)d_in[1];  // [64,1024,4] f32
    const float* subcoef = (const float*)d_in[2];  // [2] f32
    float* out = (float*)d_out;                    // scalar f32

    unsigned* dirP  = (unsigned*)d_ws;                            // 16 MB
    float* lossBuf  = (float*)((char*)d_ws + ((size_t)16 << 20)); // 64 f32

    dtw_forward_kernel<<<Bn, Tn, 0, stream>>>(preds, targs, dirP);

    const size_t ldsBytes = (size_t)WORDS_PER_BATCH * 4 + 2 * (2 * Tn) * 4; // 272 KB
    dtw_backtrace_kernel<<<Bn, Tn, ldsBytes, stream>>>(preds, targs, subcoef,
                                                       dirP, lossBuf);

    dtw_reduce_kernel<<<1, 1, 0, stream>>>(lossBuf, out);
}